// ComplicatedTransformerBlock_64742337020026
// MI455X (gfx1250) — compile-verified
//
#include <hip/hip_runtime.h>

// ComplicatedTransformerBlock, MI455X (gfx1250), fp32.
//
// The reference collapses: softmax rows sum to 1, so attn @ ones == ones;
// the reattn mix of an all-ones tensor is a per-head constant t_h; y rows are
// t_h * colsum(V) independent of sequence position; the projection output is
// one row per batch broadcast over N. Optimal kernel = 32 MB of traffic
// (~1.4 us @ 23.3 TB/s). The surviving GEMM (padded 16x1024 @ 1024x1024)
// runs on V_WMMA_F32_16X16X4_F32.

#define B_ 4
#define N_ 1024
#define E_ 1024
#define H_ 16
#define D_ 64
#define SPLIT_ 8          // split of N for the V column-sum reduction
#define EPS_ 1e-6f

typedef __attribute__((ext_vector_type(2))) float v2f;
typedef __attribute__((ext_vector_type(8))) float v8f;

// ---------------------------------------------------------------------------
// Kernel A: partial column sums of V.  partial[(b*SPLIT+s)*E + e] =
//           sum over rows [s*128, (s+1)*128) of V[b, row, e].
// Fully coalesced streaming reads of the 16 MB V tensor (non-temporal: read
// once, never reused).
// ---------------------------------------------------------------------------
__global__ void vsum_partial_kernel(const float* __restrict__ v,
                                    float* __restrict__ partial) {
    const int e = blockIdx.x * blockDim.x + threadIdx.x;  // 0..E-1
    const int s = blockIdx.y;                             // 0..SPLIT-1
    const int b = blockIdx.z;                             // 0..B-1
    const int rows = N_ / SPLIT_;
    const float* p = v + ((size_t)b * N_ + (size_t)s * rows) * E_ + e;
    float acc = 0.0f;
#pragma unroll 4
    for (int r = 0; r < rows; ++r) {
        acc += __builtin_nontemporal_load(p);
        p += E_;
    }
    partial[(b * SPLIT_ + s) * E_ + e] = acc;
}

// ---------------------------------------------------------------------------
// Kernel B: single tiny block.
//  1) t_g = colsum_g(reattn_w) * reattn_scale_g * rsqrt(mean_g(colsum^2)+eps)
//  2) Ypad[m, e] (16 x E, rows 4..15 zero) = t[e/64] * Vsum[m, e]
// Zero-padding the M dimension to 16 keeps the WMMA kernel free of any
// divergence (EXEC must be all 1s for WMMA).
// ---------------------------------------------------------------------------
__global__ void finalize_y_kernel(const float* __restrict__ partial,
                                  const float* __restrict__ reattn_w,
                                  const float* __restrict__ reattn_scale,
                                  float* __restrict__ ypad) {
    __shared__ float s_col[H_];
    __shared__ float t[H_];
    const int tid = threadIdx.x;

    if (tid < H_) {
        float s = 0.0f;
        for (int h = 0; h < H_; ++h) s += reattn_w[h * H_ + tid];
        s_col[tid] = s;
    }
    __syncthreads();
    if (tid < H_) {
        float ms = 0.0f;
        for (int g = 0; g < H_; ++g) ms += s_col[g] * s_col[g];
        ms *= (1.0f / (float)H_);
        t[tid] = s_col[tid] * reattn_scale[tid] * rsqrtf(ms + EPS_);
    }
    __syncthreads();

    for (int idx = tid; idx < 16 * E_; idx += blockDim.x) {
        const int m = idx / E_;
        const int e = idx - m * E_;
        float val = 0.0f;
        if (m < B_) {
            float acc = 0.0f;
#pragma unroll
            for (int s = 0; s < SPLIT_; ++s)
                acc += partial[(m * SPLIT_ + s) * E_ + e];
            val = t[e / D_] * acc;
        }
        ypad[idx] = val;
    }
}

// ---------------------------------------------------------------------------
// Kernel C: R = Ypad(16 x 1024) @ proj_w^T(1024 x 1024) + proj_b, via
// V_WMMA_F32_16X16X4_F32.  One wave32 per 16-column tile of R (64 waves),
// K marched in steps of 4.
// Fragment layouts per CDNA5 ISA 7.12.2:
//   A 16x4 f32 : lane m=lane&15 holds {Y[m][k0+kh], Y[m][k0+kh+1]},
//                kh = 0 for lanes 0-15, 2 for lanes 16-31.
//   B 4x16 f32 : lane col=n0+(lane&15) holds {Wt[k0+kh][col], Wt[k0+kh+1][col]}
//                where Wt[k][j] = proj_w[j*E + k].
//   C/D 16x16  : lane l<16 VGPR r holds D[r][l]  -> rows 0..3 are the batches.
// ---------------------------------------------------------------------------
__global__ void project_wmma_kernel(const float* __restrict__ ypad,
                                    const float* __restrict__ proj_w,
                                    const float* __restrict__ proj_b,
                                    float* __restrict__ R) {
    const int lane = threadIdx.x;            // 0..31 (one wave)
    const int n0   = blockIdx.x * 16;        // output column tile
    const int m    = lane & 15;
    const int kh   = (lane >> 4) << 1;       // 0 or 2

    const float* arow = ypad + (size_t)m * E_;           // A: row m of Ypad
    const float* brow = proj_w + (size_t)(n0 + m) * E_;  // B: column n0+m of Wt

    v8f c = {};
#pragma unroll 4
    for (int k0 = 0; k0 < E_; k0 += 4) {
        v2f a  = *(const v2f*)(arow + k0 + kh);
        v2f bb = *(const v2f*)(brow + k0 + kh);
        // (neg_a, A, neg_b, B, c_mod, C, reuse_a, reuse_b)
        c = __builtin_amdgcn_wmma_f32_16x16x4_f32(
                false, a, false, bb, (short)0, c, false, false);
    }

    if (lane < 16) {
        const float bias = proj_b[n0 + lane];
#pragma unroll
        for (int b = 0; b < B_; ++b)
            R[b * E_ + n0 + lane] = c[b] + bias;   // VGPR b = row b, col = lane
    }
}

// ---------------------------------------------------------------------------
// Kernel D: broadcast each batch's row over all N sequence positions.
// out[b, i, e] = R[b, e].  16 MB of coalesced NT stores; R stays in L2.
// ---------------------------------------------------------------------------
__global__ void broadcast_out_kernel(const float* __restrict__ R,
                                     float* __restrict__ out) {
    const size_t idx = (size_t)blockIdx.x * blockDim.x + threadIdx.x; // < B*N*E
    const int e = (int)(idx & (size_t)(E_ - 1));
    const int b = (int)(idx >> 20);                 // N_*E_ == 1<<20
    __builtin_nontemporal_store(R[b * E_ + e], out + idx);
}

// ---------------------------------------------------------------------------
extern "C" void kernel_launch(void* const* d_in, const int* in_sizes, int n_in,
                              void* d_out, int out_size, void* d_ws, size_t ws_size,
                              hipStream_t stream) {
    // setup_inputs order: q(0) k(1) v(2) qnorm(3) knorm(4) reattn_w(5)
    //                     reattn_norm_scale(6) proj_w(7) proj_b(8)
    // q, k, qnorm, knorm are mathematically dead (softmax rows sum to 1).
    const float* v            = (const float*)d_in[2];
    const float* reattn_w     = (const float*)d_in[5];
    const float* reattn_scale = (const float*)d_in[6];
    const float* proj_w       = (const float*)d_in[7];
    const float* proj_b       = (const float*)d_in[8];
    float* out = (float*)d_out;

    // Workspace layout (floats): partial | Ypad(16xE) | R(BxE)  (~212 KB)
    float* ws      = (float*)d_ws;
    float* partial = ws;                               // B*SPLIT*E = 32768
    float* ypad    = partial + B_ * SPLIT_ * E_;       // 16*E     = 16384
    float* R       = ypad + 16 * E_;                   // B*E      =  4096

    vsum_partial_kernel<<<dim3(E_ / 256, SPLIT_, B_), 256, 0, stream>>>(v, partial);
    finalize_y_kernel<<<1, 256, 0, stream>>>(partial, reattn_w, reattn_scale, ypad);
    project_wmma_kernel<<<E_ / 16, 32, 0, stream>>>(ypad, proj_w, proj_b, R);
    broadcast_out_kernel<<<(B_ * N_ * E_) / 256, 256, 0, stream>>>(R, out);
}